// loss_generator_64175401337456
// MI455X (gfx1250) — compile-verified
//
#include <hip/hip_runtime.h>
#include <hip/hip_bf16.h>

// Tile geometry: 128-wide x 16-tall output tile per 256-thread block.
#define TX 128
#define TY 16
#define HALO_H (TY + 2)          // 18 rows of halo
#define LDS_W 132                // TX+2 padded to multiple of 4 floats (16B)
#define CHUNKS_PER_ROW (LDS_W/4) // 33 x 16B chunks per halo row
#define NCHUNK (HALO_H * CHUNKS_PER_ROW) // 594

#if __has_builtin(__builtin_amdgcn_global_load_async_to_lds_b128) && \
    __has_builtin(__builtin_amdgcn_s_wait_asynccnt)
#define USE_ASYNC_LDS 1
#else
#define USE_ASYNC_LDS 0
#endif

typedef int v4i __attribute__((vector_size(16)));
typedef __attribute__((address_space(1))) v4i* gptr_v4i;
typedef __attribute__((address_space(3))) v4i* lptr_v4i;

__device__ __forceinline__ void copy16_to_lds(const float* gsrc, float* ldst) {
#if USE_ASYNC_LDS
    // gfx1250 async DMA: global -> LDS, 16 bytes per lane, tracked by ASYNCcnt.
    __builtin_amdgcn_global_load_async_to_lds_b128(
        (gptr_v4i)gsrc,
        (lptr_v4i)ldst,
        /*ioffset=*/0, /*cpol=*/0);
#else
    const float4 v = *(const float4*)gsrc;
    *(float4*)ldst = v;
#endif
}

__global__ __launch_bounds__(256)
void pde_residual_kernel(const float* __restrict__ g,   // (4,66,2,512,512)
                         const float* __restrict__ Rp,  // scalar R
                         float* __restrict__ fu,        // (4,64,510,510)
                         float* __restrict__ fv)        // (4,64,510,510)
{
    __shared__ __attribute__((aligned(16))) float su[HALO_H * LDS_W];
    __shared__ __attribute__((aligned(16))) float sv[HALO_H * LDS_W];

    const int tid     = threadIdx.x;
    const int tile_x0 = blockIdx.x * TX;   // 0,128,256,384
    const int tile_y0 = blockIdx.y * TY;   // 0..496
    const int bz      = blockIdx.z;        // b*64 + s_out
    const int b       = bz >> 6;
    const int s       = bz & 63;
    const int t       = s + 1;             // center time slice in u_full

    const size_t SL = (size_t)512 * 512;
    // channel base pointers: element (b, t', c, y, x)
    const float* ub  = g + ((size_t)(b * 66 + t)     * 2) * SL;      // u @ t
    const float* vb  = ub + SL;                                      // v @ t
    const float* ubm = g + ((size_t)(b * 66 + t - 1) * 2) * SL;      // u @ t-1
    const float* vbm = ubm + SL;
    const float* ubp = g + ((size_t)(b * 66 + t + 1) * 2) * SL;      // u @ t+1
    const float* vbp = ubp + SL;

    // ---- Stage halo tiles (slice t) for u and v into LDS via async DMA ----
    #pragma unroll
    for (int it = 0; it < 3; ++it) {
        const int e = it * 256 + tid;
        if (e < NCHUNK) {
            const int r  = e / CHUNKS_PER_ROW;
            const int k  = e - r * CHUNKS_PER_ROW;
            const int gy = tile_y0 + r;            // global y of halo row
            const int gx = tile_x0 + 4 * k;        // global x of 4-float chunk
            if (gy < 512 && gx <= 508) {           // in-bounds 16B chunk
                const size_t gofs = (size_t)gy * 512 + gx;
                const int    lofs = r * LDS_W + 4 * k;
                copy16_to_lds(ub + gofs, &su[lofs]);
                copy16_to_lds(vb + gofs, &sv[lofs]);
            }
        }
    }
#if USE_ASYNC_LDS
    __builtin_amdgcn_s_wait_asynccnt(0);
#endif
    __syncthreads();

    // ---- Constants (match reference semantics to ~1 ulp) ----
    // DX = 20/128 = 0.15625 exact; DX^2 = 25/1024 exact. DT = 0.05.
    const float inv_dx2 = (float)(1.0 / ((double)0.15625 * 0.15625)); // 40.96
    const float inv_2dx = (float)(1.0 / (2.0 * 0.15625));             // 3.2
    const float inv_2dt = (float)(1.0 / (2.0 * (double)(float)(10.0 / 200.0)));
    const float invR    = 1.0f / Rp[0];

    const int tx   = tid & (TX - 1);   // 0..127
    const int half = tid >> 7;         // 0 or 1
    const int xo   = tile_x0 + tx;     // output column
    if (xo < 510) {
        const int lx = tx + 1;         // halo-local center column
        const size_t out_base = ((size_t)(b * 64 + s) * 510) * 510;
        #pragma unroll
        for (int j = 0; j < TY / 2; ++j) {
            const int row = j * 2 + half;
            const int yo  = tile_y0 + row;
            if (yo < 510) {
                const int    iy   = yo + 1, ix = xo + 1;
                const size_t gidx = (size_t)iy * 512 + ix;
                const int    c    = (row + 1) * LDS_W + lx;

                const float ucc = su[c],          vcc = sv[c];
                const float u_n = su[c - LDS_W],  u_s = su[c + LDS_W];
                const float u_w = su[c - 1],      u_e = su[c + 1];
                const float v_n = sv[c - LDS_W],  v_s = sv[c + LDS_W];
                const float v_w = sv[c - 1],      v_e = sv[c + 1];

                const float lap_u = (4.0f * ucc - u_n - u_s - u_w - u_e) * inv_dx2;
                const float lap_v = (4.0f * vcc - v_n - v_s - v_w - v_e) * inv_dx2;
                const float u_x = (u_s - u_n) * inv_2dx;   // d/dy-axis (ref "_ddx")
                const float v_x = (v_s - v_n) * inv_2dx;
                const float u_y = (u_e - u_w) * inv_2dx;   // d/dx-axis (ref "_ddy")
                const float v_y = (v_e - v_w) * inv_2dx;
                const float u_t = (ubp[gidx] - ubm[gidx]) * inv_2dt;
                const float v_t = (vbp[gidx] - vbm[gidx]) * inv_2dt;

                const float f_u = u_t + ucc * u_x + vcc * u_y - lap_u * invR;
                const float f_v = v_t + ucc * v_x + vcc * v_y - lap_v * invR;

                const size_t o = out_base + (size_t)yo * 510 + xo;
                __builtin_nontemporal_store(f_u, &fu[o]);  // write-once: keep out of L2
                __builtin_nontemporal_store(f_v, &fv[o]);
            }
        }
    }
}

extern "C" void kernel_launch(void* const* d_in, const int* in_sizes, int n_in,
                              void* d_out, int out_size, void* d_ws, size_t ws_size,
                              hipStream_t stream) {
    (void)in_sizes; (void)n_in; (void)out_size; (void)d_ws; (void)ws_size;
    const float* g  = (const float*)d_in[0];   // (4,66,2,512,512) f32
    const float* R  = (const float*)d_in[1];   // scalar f32
    float* fu = (float*)d_out;                                 // f_u flat
    float* fv = fu + (size_t)4 * 64 * 510 * 510;               // f_v flat

    dim3 grid((510 + TX - 1) / TX,   // 4
              (510 + TY - 1) / TY,   // 32
              4 * 64);               // (b, s_out)
    pde_residual_kernel<<<grid, 256, 0, stream>>>(g, R, fu, fv);
}